// ContextNorm_73332271612492
// MI455X (gfx1250) — compile-verified
//
#include <hip/hip_runtime.h>

// ContextNorm for MI455X (gfx1250, wave32).
// Pass 1: segment reduction (cnt, sum, sumsq per context) expressed as a
//         one-hot GEMM on V_WMMA_F32_16X16X4_F32.
// Pass 2: per-(context,dim) scale/shift (folds cnt==0 -> identity).
// Pass 3: streaming float4 FMA apply (b128 loads/stores), scale/shift in LDS.

#define EPSV   0.001f
#define NCTX   16
#define DDIM   64

typedef __attribute__((ext_vector_type(2))) float v2f;
typedef __attribute__((ext_vector_type(8))) float v8f;

#define P1_BLOCKS   512
#define P1_THREADS  256
#define WPB         (P1_THREADS / 32)       // waves per block = 8
#define PART_VALS   (NCTX + 2 * NCTX * DDIM) // 16 + 1024 + 1024 = 2064
#define PART_STRIDE 2080                     // padded record stride (floats)
#define SCALE_OFF   0
#define SHIFT_OFF   (NCTX * DDIM)            // 1024
#define PART_OFF    (2 * NCTX * DDIM)        // 2048

// -------------------------------------------------------------------------
// Pass 1: per-block partial (cnt[16], s1[16][64], s2[16][64]) via WMMA.
// A (16x4)  = one-hot of contexts  (M=context, K=sample-in-group)
// B (4x16)  = sample values        (K=sample-in-group, N=dim-in-group)
// D (16x16) accumulates segment sums. 4 dim-groups cover D=64.
// Operand layouts per CDNA5 ISA 7.12.2:
//   A: lane m=lane&15, VGPR0 holds K=0 (lanes0-15) / K=2 (lanes16-31), VGPR1 K=1/K=3
//   B: lane n=lane&15, VGPR0 holds K=0 (lanes0-15) / K=2 (lanes16-31), VGPR1 K=1/K=3
//   C/D: lane n=lane&15, VGPR r holds M=r (lanes0-15) / M=8+r (lanes16-31)
// -------------------------------------------------------------------------
__global__ __launch_bounds__(P1_THREADS)
void cn_reduce(const float* __restrict__ x, const int* __restrict__ ctx,
               float* __restrict__ part, int n)
{
    __shared__ float smem[WPB][PART_STRIDE];

    const int lane = threadIdx.x & 31;
    const int wave = threadIdx.x >> 5;
    const int half = lane >> 4;     // 0: lanes 0-15, 1: lanes 16-31
    const int m    = lane & 15;     // A-row (context) / B-col (dim) per lane

    const unsigned gwave   = blockIdx.x * WPB + wave;
    const unsigned nwaves  = P1_BLOCKS * WPB;
    const unsigned ngroups = ((unsigned)n) >> 2;   // groups of 4 samples

    v8f s1[4] = {};   // segment sums, dims [16g, 16g+16)
    v8f s2[4] = {};   // segment sums of squares
    v8f cnt   = {};   // counts (replicated across N columns)

    v2f bones; bones[0] = 1.0f; bones[1] = 1.0f;

    for (unsigned g = gwave; g < ngroups; g += nwaves) {
        const unsigned s0 = g << 2;
        const unsigned sE = s0 + (half ? 2u : 0u);  // K=0 / K=2 sample
        const unsigned sO = s0 + (half ? 3u : 1u);  // K=1 / K=3 sample

        const int cE = ctx[sE];
        const int cO = ctx[sO];
        v2f a;
        a[0] = (cE == m) ? 1.0f : 0.0f;
        a[1] = (cO == m) ? 1.0f : 0.0f;

        cnt = __builtin_amdgcn_wmma_f32_16x16x4_f32(
                  false, a, false, bones, (short)0, cnt, false, false);

        const float* rowE = x + (size_t)sE * DDIM + m;
        const float* rowO = x + (size_t)sO * DDIM + m;
#pragma unroll
        for (int grp = 0; grp < 4; ++grp) {
            v2f b, b2;
            b[0]  = rowE[grp * 16];
            b[1]  = rowO[grp * 16];
            b2[0] = b[0] * b[0];
            b2[1] = b[1] * b[1];
            s1[grp] = __builtin_amdgcn_wmma_f32_16x16x4_f32(
                          false, a, false, b, (short)0, s1[grp], false, false);
            s2[grp] = __builtin_amdgcn_wmma_f32_16x16x4_f32(
                          false, a, false, b2, (short)0, s2[grp], false, false);
        }
    }

    // Spill per-wave partials to this wave's private LDS slab.
#pragma unroll
    for (int r = 0; r < 8; ++r) {
        const int c = r + half * 8;
        if (m == 0) smem[wave][c] = cnt[r];   // every N-column of cnt is equal
#pragma unroll
        for (int grp = 0; grp < 4; ++grp) {
            const int d = grp * 16 + m;
            smem[wave][NCTX + c * DDIM + d]                = s1[grp][r];
            smem[wave][NCTX + NCTX * DDIM + c * DDIM + d]  = s2[grp][r];
        }
    }
    __syncthreads();

    // Tail samples (n % 4), handled once, scalar (n = 1e6 -> empty).
    if (blockIdx.x == 0 && threadIdx.x == 0) {
        for (unsigned s = (ngroups << 2); s < (unsigned)n; ++s) {
            const int c = ctx[s];
            smem[0][c] += 1.0f;
            for (int d = 0; d < DDIM; ++d) {
                const float v = x[(size_t)s * DDIM + d];
                smem[0][NCTX + c * DDIM + d]               += v;
                smem[0][NCTX + NCTX * DDIM + c * DDIM + d] += v * v;
            }
        }
    }
    __syncthreads();

    // Fixed-order cross-wave reduction -> one partial record per block.
    for (int i = threadIdx.x; i < PART_VALS; i += P1_THREADS) {
        float acc = 0.0f;
#pragma unroll
        for (int w = 0; w < WPB; ++w) acc += smem[w][i];
        part[(size_t)blockIdx.x * PART_STRIDE + i] = acc;
    }
}

// -------------------------------------------------------------------------
// Pass 2: reduce block partials, compute scale/shift per (context, dim).
// cnt==0 -> scale=1, shift=0 reproduces jnp.where(cnt>0, y, samples).
// -------------------------------------------------------------------------
__global__ __launch_bounds__(256)
void cn_stats(const float* __restrict__ part,
              const float* __restrict__ gamma, const float* __restrict__ beta,
              const float* __restrict__ priors,
              float* __restrict__ scale, float* __restrict__ shift)
{
    const int t = blockIdx.x * blockDim.x + threadIdx.x;
    if (t >= NCTX * DDIM) return;
    const int c = t >> 6;
    const int d = t & (DDIM - 1);

    float cnt = 0.0f, sa = 0.0f, sb = 0.0f;
    for (int blk = 0; blk < P1_BLOCKS; ++blk) {
        const float* p = part + (size_t)blk * PART_STRIDE;
        cnt += p[c];
        sa  += p[NCTX + c * DDIM + d];
        sb  += p[NCTX + NCTX * DDIM + c * DDIM + d];
    }

    float sc, sh;
    if (cnt > 0.0f) {
        const float inv  = 1.0f / cnt;
        const float mean = sa * inv;
        const float var  = sb * inv - mean * mean;
        const float istd = rsqrtf(var + EPSV);
        const float ps   = rsqrtf(priors[c]);
        const float g    = gamma[c * DDIM + d];
        sc = g * istd * ps;
        sh = (beta[c * DDIM + d] - g * mean * istd) * ps;
    } else {
        sc = 1.0f;
        sh = 0.0f;
    }
    scale[t] = sc;
    shift[t] = sh;
}

// -------------------------------------------------------------------------
// Pass 3: y = x * scale[c] + shift[c], pure float4 stream (b128 ld/st).
// -------------------------------------------------------------------------
__global__ __launch_bounds__(256)
void cn_apply(const float* __restrict__ x, const int* __restrict__ ctx,
              const float* __restrict__ scale, const float* __restrict__ shift,
              float* __restrict__ y, int n)
{
    __shared__ __align__(16) float s_scale[NCTX * DDIM];
    __shared__ __align__(16) float s_shift[NCTX * DDIM];
    for (int i = threadIdx.x; i < NCTX * DDIM; i += blockDim.x) {
        s_scale[i] = scale[i];
        s_shift[i] = shift[i];
    }
    __syncthreads();

    const float4* x4 = (const float4*)x;
    float4*       y4 = (float4*)y;
    const unsigned total  = (unsigned)n * (DDIM / 4);   // n*16 float4s
    const unsigned stride = gridDim.x * blockDim.x;

    for (unsigned i = blockIdx.x * blockDim.x + threadIdx.x; i < total; i += stride) {
        const unsigned s = i >> 4;        // sample index (16 float4s per row)
        const unsigned q = i & 15;        // float4 within row
        const int      c = ctx[s];        // 16 lanes share addr -> coalesced

        const float4 v  = x4[i];
        const float4 sc = ((const float4*)s_scale)[c * (DDIM / 4) + q];
        const float4 sh = ((const float4*)s_shift)[c * (DDIM / 4) + q];

        float4 o;
        o.x = fmaf(v.x, sc.x, sh.x);
        o.y = fmaf(v.y, sc.y, sh.y);
        o.z = fmaf(v.z, sc.z, sh.z);
        o.w = fmaf(v.w, sc.w, sh.w);
        y4[i] = o;
    }
}

extern "C" void kernel_launch(void* const* d_in, const int* in_sizes, int n_in,
                              void* d_out, int out_size, void* d_ws, size_t ws_size,
                              hipStream_t stream)
{
    const float* samples = (const float*)d_in[0];
    const int*   ctx     = (const int*)  d_in[1];
    const float* gamma   = (const float*)d_in[2];
    const float* beta    = (const float*)d_in[3];
    const float* priors  = (const float*)d_in[4];
    float*       out     = (float*)d_out;

    const int n = in_sizes[0] / DDIM;    // number of samples

    float* ws    = (float*)d_ws;
    float* scale = ws + SCALE_OFF;                 // 1024 floats
    float* shift = ws + SHIFT_OFF;                 // 1024 floats
    float* part  = ws + PART_OFF;                  // P1_BLOCKS * PART_STRIDE floats

    cn_reduce<<<P1_BLOCKS, P1_THREADS, 0, stream>>>(samples, ctx, part, n);
    cn_stats <<<(NCTX * DDIM + 255) / 256, 256, 0, stream>>>(part, gamma, beta,
                                                             priors, scale, shift);
    cn_apply <<<4096, 256, 0, stream>>>(samples, ctx, scale, shift, out, n);
}